// HungarianMatcher_70506183131643
// MI455X (gfx1250) — compile-verified
//
#include <hip/hip_runtime.h>
#include <math.h>

// ---------------------------------------------------------------------------
// HungarianMatcher for MI455X (gfx1250, wave32)
//   Phase 1: batched A·Bᵀ (16 x [64x50176]) via V_WMMA_F32_16X16X4_F32,
//            split-K over D into 16 slices for bandwidth saturation,
//            deterministic partial-sum reduction (no float atomics).
//   Phase 2: exact Jonker-Volgenant LSA, one wave32 per batch.
// ---------------------------------------------------------------------------

typedef float v2f __attribute__((ext_vector_type(2)));
typedef float v8f __attribute__((ext_vector_type(8)));

#define BATCH   16
#define KDIM    64          // k group masks (cost matrix is 64x64)
#define DDIM    50176       // 224*224
#define KC      64          // D-chunk staged in LDS per iteration
#define LDA     68          // padded LDS row stride (floats): conflict-free
#define NSLICE  16          // split-K slices over D
#define CHUNKS_PER_SLICE ((DDIM / KC) / NSLICE)   // 784/16 = 49

// ---------------------------------------------------------------------------
// Kernel 1: partial dot-product tiles + partial squared norms for one
// (batch, D-slice). grid = BATCH*NSLICE blocks of 256 threads (8 waves).
// Each wave owns two 16x16 tiles of the 64x64 product; each input element is
// read from HBM exactly once.
// ---------------------------------------------------------------------------
__global__ __launch_bounds__(256) void cost_partial_wmma(
    const float* __restrict__ A,      // [16,64,50176] outputs
    const float* __restrict__ T,      // [16,64,50176] targets
    float* __restrict__ ab_part,      // [16][NSLICE][64*64]
    float* __restrict__ nm_part) {    // [16][NSLICE][128] (a-norms then b-norms)
  __shared__ float lds_a[KDIM * LDA];
  __shared__ float lds_b[KDIM * LDA];
  __shared__ float red[512];

  const int t    = threadIdx.x;
  const int b    = blockIdx.x >> 4;
  const int sl   = blockIdx.x & 15;
  const int row  = t >> 2;          // 0..63 : row this thread stages
  const int seg  = t & 3;           // 16-float segment within the KC chunk
  const int wave = t >> 5;
  const int lane = t & 31;
  const int m15  = lane & 15;
  const int kb   = (lane >> 4) * 2; // ISA A/B frag: lanes 16-31 hold K=2,3

  const int tile0 = wave * 2, tile1 = wave * 2 + 1;
  const int ti0 = tile0 >> 2, tj0 = tile0 & 3;
  const int ti1 = tile1 >> 2, tj1 = tile1 & 3;

  const int ch0 = sl * CHUNKS_PER_SLICE;
  const float* ga = A + ((size_t)(b * KDIM + row)) * DDIM + seg * 16;
  const float* gb = T + ((size_t)(b * KDIM + row)) * DDIM + seg * 16;
  float* la = &lds_a[row * LDA + seg * 16];
  float* lb = &lds_b[row * LDA + seg * 16];

  v8f acc0 = {}; v8f acc1 = {};
  float sa = 0.f, sb = 0.f;

  for (int c = 0; c < CHUNKS_PER_SLICE; ++c) {
    const size_t off = (size_t)(ch0 + c) * KC;
    #pragma unroll
    for (int q = 0; q < 4; ++q) {
      float4 xa = *(const float4*)(ga + off + q * 4);
      float4 xb = *(const float4*)(gb + off + q * 4);
      sa += xa.x * xa.x + xa.y * xa.y + xa.z * xa.z + xa.w * xa.w;
      sb += xb.x * xb.x + xb.y * xb.y + xb.z * xb.z + xb.w * xb.w;
      *(float4*)(la + q * 4) = xa;
      *(float4*)(lb + q * 4) = xb;
    }
    __syncthreads();

    const float* pa0 = &lds_a[(ti0 * 16 + m15) * LDA + kb];
    const float* pb0 = &lds_b[(tj0 * 16 + m15) * LDA + kb];
    const float* pa1 = &lds_a[(ti1 * 16 + m15) * LDA + kb];
    const float* pb1 = &lds_b[(tj1 * 16 + m15) * LDA + kb];
    #pragma unroll
    for (int kk = 0; kk < KC; kk += 4) {
      v2f av0 = *(const v2f*)(pa0 + kk);
      v2f bv0 = *(const v2f*)(pb0 + kk);
      acc0 = __builtin_amdgcn_wmma_f32_16x16x4_f32(
          false, av0, false, bv0, (short)0, acc0, false, false);
      v2f av1 = *(const v2f*)(pa1 + kk);
      v2f bv1 = *(const v2f*)(pb1 + kk);
      acc1 = __builtin_amdgcn_wmma_f32_16x16x4_f32(
          false, av1, false, bv1, (short)0, acc1, false, false);
    }
    __syncthreads();
  }

  // reduce per-row partial squared norms (4 threads per row)
  red[t] = sa; red[256 + t] = sb;
  __syncthreads();
  float* nm = nm_part + (size_t)(b * NSLICE + sl) * 128;
  if (t < 64) {
    nm[t]      = red[4 * t] + red[4 * t + 1] + red[4 * t + 2] + red[4 * t + 3];
    nm[64 + t] = red[256 + 4 * t] + red[256 + 4 * t + 1] +
                 red[256 + 4 * t + 2] + red[256 + 4 * t + 3];
  }

  // store partial product tiles (C/D layout: VGPR r, lane L -> M=r+8*(L>>4), N=L&15)
  float* abp = ab_part + (size_t)(b * NSLICE + sl) * (KDIM * KDIM);
  #pragma unroll
  for (int r = 0; r < 8; ++r) {
    {
      int M = ti0 * 16 + r + 8 * (lane >> 4);
      int N = tj0 * 16 + m15;
      abp[M * KDIM + N] = acc0[r];
    }
    {
      int M = ti1 * 16 + r + 8 * (lane >> 4);
      int N = tj1 * 16 + m15;
      abp[M * KDIM + N] = acc1[r];
    }
  }
}

// ---------------------------------------------------------------------------
// Kernel 2: deterministic slice reduction + cost = sqrt(max(a2+b2-2ab,0)).
// grid = BATCH blocks of 256 threads.
// ---------------------------------------------------------------------------
__global__ __launch_bounds__(256) void cost_finalize(
    const float* __restrict__ ab_part,
    const float* __restrict__ nm_part,
    float* __restrict__ cost) {       // [16][64*64]
  const int b = blockIdx.x, t = threadIdx.x;
  __shared__ float norms[128];
  if (t < 128) {
    float s = 0.f;
    for (int sl = 0; sl < NSLICE; ++sl)
      s += nm_part[(size_t)(b * NSLICE + sl) * 128 + t];
    norms[t] = s;
  }
  __syncthreads();
  const float* base = ab_part + (size_t)b * NSLICE * (KDIM * KDIM);
  for (int e = t; e < KDIM * KDIM; e += 256) {
    float ab = 0.f;
    for (int sl = 0; sl < NSLICE; ++sl) ab += base[(size_t)sl * KDIM * KDIM + e];
    int M = e >> 6, N = e & 63;
    float c = norms[M] + norms[64 + N] - 2.f * ab;
    cost[(size_t)b * KDIM * KDIM + e] = sqrtf(fmaxf(c, 0.f));
  }
}

// ---------------------------------------------------------------------------
// Kernel 3: exact Jonker-Volgenant LSA (e-maxx formulation, 1-indexed with
// dummy col 0), one wave32 per batch. Each lane owns columns lane+1, lane+33.
// Argmin over free columns via shfl_xor butterfly; ties -> lowest index,
// matching jnp.argmin.
// ---------------------------------------------------------------------------
__global__ __launch_bounds__(32) void lsa_wave(
    const float* __restrict__ costAll,
    int* __restrict__ out1,           // [16][64] arange
    int* __restrict__ out2) {         // [16][64] col_of_row
  const int b = blockIdx.x;
  const int lane = threadIdx.x;
  __shared__ float cst[KDIM * KDIM];
  __shared__ float u[KDIM + 1];
  __shared__ int   p[KDIM + 1];
  __shared__ int   way[KDIM + 1];

  const float* src = costAll + (size_t)b * KDIM * KDIM;
  for (int idx = lane; idx < KDIM * KDIM; idx += 32) cst[idx] = src[idx];
  for (int idx = lane; idx <= KDIM; idx += 32) { u[idx] = 0.f; p[idx] = 0; way[idx] = 0; }
  __syncthreads();

  const int col0 = lane + 1;
  const int col1 = lane + 33;
  float v0 = 0.f, v1 = 0.f;
  const float INF = 1e30f;

  for (int i = 1; i <= KDIM; ++i) {
    if (lane == 0) { p[0] = i; way[0] = 0; }
    way[col0] = 0; way[col1] = 0;
    float minv0 = INF, minv1 = INF;
    bool used0 = false, used1 = false, usedZ = false;
    __syncthreads();

    int j0 = 0;
    while (true) {
      if (j0 == 0)          usedZ = true;
      else if (j0 == col0)  used0 = true;
      else if (j0 == col1)  used1 = true;

      int   i0  = p[j0];                 // j0 uniform -> LDS broadcast
      float ui0 = u[i0];
      const float* crow = &cst[(i0 - 1) * KDIM];
      float cur0 = crow[col0 - 1] - ui0 - v0;
      float cur1 = crow[col1 - 1] - ui0 - v1;
      if (!used0 && cur0 < minv0) { minv0 = cur0; way[col0] = j0; }
      if (!used1 && cur1 < minv1) { minv1 = cur1; way[col1] = j0; }

      // argmin over free columns, 0-based candidate index = col-1
      float c0 = used0 ? INF : minv0;
      float c1 = used1 ? INF : minv1;
      float bv; int bi;
      if (c1 < c0) { bv = c1; bi = col1 - 1; } else { bv = c0; bi = col0 - 1; }
      #pragma unroll
      for (int off = 16; off >= 1; off >>= 1) {
        float ov = __shfl_xor(bv, off, 32);
        int   oi = __shfl_xor(bi, off, 32);
        if (ov < bv || (ov == bv && oi < bi)) { bv = ov; bi = oi; }
      }
      float delta = bv;
      int   j1    = bi + 1;

      // u[p[j]] += delta for all used j (rows are distinct: p is a matching)
      if (lane == 0 && usedZ) u[p[0]] += delta;
      if (used0) { u[p[col0]] += delta; v0 -= delta; } else { minv0 -= delta; }
      if (used1) { u[p[col1]] += delta; v1 -= delta; } else { minv1 -= delta; }
      __syncthreads();

      j0 = j1;
      if (p[j0] == 0) break;
    }

    __syncthreads();
    if (lane == 0) {                     // augment along alternating path
      int jj = j0;
      while (jj != 0) { int jn = way[jj]; p[jj] = p[jn]; jj = jn; }
    }
    __syncthreads();
  }

  out1[b * KDIM + lane]      = lane;
  out1[b * KDIM + lane + 32] = lane + 32;
  int r0 = p[col0] - 1;
  int r1 = p[col1] - 1;
  out2[b * KDIM + r0] = col0 - 1;
  out2[b * KDIM + r1] = col1 - 1;
}

// ---------------------------------------------------------------------------
extern "C" void kernel_launch(void* const* d_in, const int* in_sizes, int n_in,
                              void* d_out, int out_size, void* d_ws, size_t ws_size,
                              hipStream_t stream) {
  const float* A = (const float*)d_in[0];   // outputs [16,64,50176] f32
  const float* T = (const float*)d_in[1];   // targets [16,64,50176] f32

  char* ws = (char*)d_ws;
  const size_t ab_bytes = (size_t)BATCH * NSLICE * KDIM * KDIM * sizeof(float); // 4 MiB
  const size_t nm_bytes = (size_t)BATCH * NSLICE * 128 * sizeof(float);          // 128 KiB
  float* ab_part = (float*)ws;
  float* nm_part = (float*)(ws + ab_bytes);
  float* cost    = (float*)(ws + ab_bytes + nm_bytes);                           // 256 KiB

  int* out1 = (int*)d_out;           // row indices (arange), int32
  int* out2 = out1 + BATCH * KDIM;   // matched target index per pred, int32

  cost_partial_wmma<<<dim3(BATCH * NSLICE), dim3(256), 0, stream>>>(A, T, ab_part, nm_part);
  cost_finalize<<<dim3(BATCH), dim3(256), 0, stream>>>(ab_part, nm_part, cost);
  lsa_wave<<<dim3(BATCH), dim3(32), 0, stream>>>(cost, out1, out2);
}